// S4Cell_8546984919597
// MI455X (gfx1250) — compile-verified
//
#include <hip/hip_runtime.h>
#include <math.h>
#include <cstdint>

#define Lseq 2048
#define Hch  512
#define Nst  64

typedef __attribute__((ext_vector_type(2))) float v2f;
typedef __attribute__((ext_vector_type(8))) float v8f;
typedef __attribute__((ext_vector_type(4))) unsigned int u32x4;
typedef __attribute__((ext_vector_type(4))) int i32x4;
typedef __attribute__((ext_vector_type(8))) int i32x8;

// ---------------------------------------------------------------------------
// Kernel 1: per channel h:
//   roots[l] = c2(l) * (k00 - k01 * k10 / (1 + k11))   (Cauchy sums over N=64)
//   kern[h,:] = Re( IFFT_2048(roots) )                  (radix-2 DIT in LDS)
// ---------------------------------------------------------------------------
__global__ __launch_bounds__(256) void s4_kern_ifft(
    const float* __restrict__ lamRg, const float* __restrict__ lamIg,
    const float* __restrict__ pRg,   const float* __restrict__ pIg,
    const float* __restrict__ bRg,   const float* __restrict__ bIg,
    const float* __restrict__ cg,    const float* __restrict__ logstep,
    float* __restrict__ kernG)
{
    __shared__ float xr[Lseq];
    __shared__ float xi[Lseq];
    __shared__ float sLr[Nst], sLi[Nst];
    __shared__ float w00r[Nst], w00i[Nst], w01r[Nst], w01i[Nst];
    __shared__ float w10r[Nst], w10i[Nst], w11r[Nst], w11i[Nst];

    const int h   = blockIdx.x;
    const int tid = threadIdx.x;

    // Stage per-channel weights into LDS (cc = conj(c), qc = conj(p)).
    if (tid < Nst) {
        const int n  = tid;
        const int ix = h * Nst + n;
        float lr = fminf(lamRg[ix], -1.0e-4f);   // clip(lambda_real, max=-1e-4)
        float li = lamIg[ix];
        float pr = pRg[ix], pii = pIg[ix];
        float br = bRg[ix], bi  = bIg[ix];
        float c0 = cg[2 * ix + 0], c1 = cg[2 * ix + 1];
        float ccr = c0, cci = -c1;               // conj(c0 + i c1)
        float qr  = pr, qi  = -pii;              // conj(p)
        sLr[n] = lr;  sLi[n] = li;
        w00r[n] = ccr * br - cci * bi;   w00i[n] = ccr * bi  + cci * br;   // cc*bc
        w01r[n] = ccr * pr - cci * pii;  w01i[n] = ccr * pii + cci * pr;   // cc*pc
        w10r[n] = qr  * br - qi  * bi;   w10i[n] = qr  * bi  + qi  * br;   // qc*bc
        w11r[n] = qr  * pr - qi  * pii;  w11i[n] = qr  * pii + qi  * pr;   // qc*pc
    }
    __syncthreads();

    const float step = expf(logstep[h]);
    const float tos  = 2.0f / step;
    const float invL = 1.0f / (float)Lseq;
    const float TWO_PI = 6.28318530717958647692f;

    // Each thread evaluates 8 frequency bins.
    for (int qq = 0; qq < 8; ++qq) {
        const int l = qq * 256 + tid;
        const float th  = TWO_PI * (float)l * invL;
        const float omr = cosf(th);
        const float omi = -sinf(th);                  // omega = exp(-i*th)
        const float dr  = 1.0f + omr, di = omi;       // 1 + omega
        const float inv = 1.0f / (dr * dr + di * di);
        const float nr  = 1.0f - omr, ni = -omi;      // 1 - omega
        const float gr  = tos * ((nr * dr + ni * di) * inv);
        const float gi  = tos * ((ni * dr - nr * di) * inv);
        const float c2r = 2.0f * dr * inv;            // 2 / (1 + omega)
        const float c2i = -2.0f * di * inv;

        float k00r = 0.f, k00i = 0.f, k01r = 0.f, k01i = 0.f;
        float k10r = 0.f, k10i = 0.f, k11r = 0.f, k11i = 0.f;
        #pragma unroll 4
        for (int n = 0; n < Nst; ++n) {
            const float ar = gr - sLr[n];
            const float ai = gi - sLi[n];
            const float rq = 1.0f / (ar * ar + ai * ai);
            const float er = ar * rq;
            const float ei = -ai * rq;                // 1/(g - lambda)
            k00r += w00r[n] * er - w00i[n] * ei;  k00i += w00r[n] * ei + w00i[n] * er;
            k01r += w01r[n] * er - w01i[n] * ei;  k01i += w01r[n] * ei + w01i[n] * er;
            k10r += w10r[n] * er - w10i[n] * ei;  k10i += w10r[n] * ei + w10i[n] * er;
            k11r += w11r[n] * er - w11i[n] * ei;  k11i += w11r[n] * ei + w11i[n] * er;
        }
        // roots = c2 * (k00 - k01 * k10 / (1 + k11))
        const float wr2 = 1.0f + k11r, wi2 = k11i;
        const float iw  = 1.0f / (wr2 * wr2 + wi2 * wi2);
        const float hr  = k01r * k10r - k01i * k10i;
        const float hi  = k01r * k10i + k01i * k10r;
        const float tr  = (hr * wr2 + hi * wi2) * iw;
        const float ti  = (hi * wr2 - hr * wi2) * iw;
        const float rr  = k00r - tr, ri = k00i - ti;
        const float Rr  = c2r * rr - c2i * ri;
        const float Ri  = c2r * ri + c2i * rr;
        // Store bit-reversed, folding in the 1/L IFFT scale.
        const int rev = (int)(__brev((unsigned)l) >> 21);   // 11-bit reversal
        xr[rev] = Rr * invL;
        xi[rev] = Ri * invL;
    }
    __syncthreads();

    // 11 radix-2 DIT stages, IFFT twiddle sign (+i).
    for (int s = 1; s <= 11; ++s) {
        const int half = 1 << (s - 1);
        const int len  = 1 << s;
        for (int b = tid; b < (Lseq / 2); b += 256) {
            const int grp = b >> (s - 1);
            const int pos = b & (half - 1);
            const int i0  = grp * len + pos;
            const int i1  = i0 + half;
            const float ang = TWO_PI * (float)pos / (float)len;
            const float twr = cosf(ang);
            const float twi = sinf(ang);
            const float x1r = xr[i1], x1i = xi[i1];
            const float tr  = twr * x1r - twi * x1i;
            const float ti  = twr * x1i + twi * x1r;
            const float x0r = xr[i0], x0i = xi[i0];
            xr[i1] = x0r - tr;  xi[i1] = x0i - ti;
            xr[i0] = x0r + tr;  xi[i0] = x0i + ti;
        }
        __syncthreads();
    }

    // kern = real part of the IFFT.
    for (int qq = 0; qq < 8; ++qq) {
        const int s = qq * 256 + tid;
        kernG[h * Lseq + s] = xr[s];
    }
}

// ---------------------------------------------------------------------------
// TDM: issue one tensor_load_to_lds for a 2D tile (tile0 elems/row, tile1
// rows, row stride stride0 elems, 4-byte elements) compacted into LDS.
// ---------------------------------------------------------------------------
__device__ __forceinline__ void tdm_load_f32_tile(
    uint32_t lds_off, const void* gaddr,
    uint32_t tdim0, uint32_t tdim1, uint32_t tile0, uint32_t tile1,
    uint32_t stride0)
{
    const uint64_t ga = (uint64_t)(uintptr_t)gaddr;
    u32x4 g0;
    g0[0] = 1u;                                        // count=1, no gather mode
    g0[1] = lds_off;                                   // lds_addr (bytes)
    g0[2] = (uint32_t)ga;                              // global_addr[31:0]
    g0[3] = (uint32_t)((ga >> 32) & 0x1FFFFFFu)        // global_addr[56:32]
            | (2u << 30);                              // type = 2 ("image")
    i32x8 g1;
    g1[0] = (int)(2u << 16);                           // data_size = 4 bytes
    g1[1] = (int)((tdim0 & 0xFFFFu) << 16);            // tensor_dim0[15:0] @ 63:48
    g1[2] = (int)(((tdim0 >> 16) & 0xFFFFu)            // tensor_dim0[31:16]
            | ((tdim1 & 0xFFFFu) << 16));              // tensor_dim1[15:0]
    g1[3] = (int)(((tdim1 >> 16) & 0xFFFFu)            // tensor_dim1[31:16]
            | ((tile0 & 0xFFFFu) << 16));              // tile_dim0
    g1[4] = (int)(tile1 & 0xFFFFu);                    // tile_dim1 (tile_dim2 = 0)
    g1[5] = (int)stride0;                              // tensor_dim0_stride[31:0]
    g1[6] = 0;                                         // stride0 hi / dim1_stride lo
    g1[7] = 0;
    const i32x4 z4 = {0, 0, 0, 0};
#if __has_include(<hip/amd_detail/amd_gfx1250_TDM.h>)
    const i32x8 z8 = {0, 0, 0, 0, 0, 0, 0, 0};
    __builtin_amdgcn_tensor_load_to_lds(g0, g1, z4, z4, z8, 0);
#else
    __builtin_amdgcn_tensor_load_to_lds(g0, g1, z4, z4, 0);
#endif
}

// ---------------------------------------------------------------------------
// Kernel 2: causal convolution y[t] = sum_{s<=t} kern[s] * u[t-s]  + d*u[t]
// as a block-Toeplitz matmul on V_WMMA_F32_16X16X4_F32:
//   Y_I(m,n) = sum_d A_d(m,:) . U_{16I+n-d}(:)   , A_d[m,r] = kern[16d+m-r]
// Wave I (of 8) owns output tile t in [256I, 256I+256).
// Staging global->LDS is done by the Tensor Data Mover (wave 0 issues).
// ---------------------------------------------------------------------------
__global__ __launch_bounds__(256) void s4_conv_wmma(
    const float* __restrict__ u, const float* __restrict__ kernG,
    const float* __restrict__ dvec, float* __restrict__ out)
{
    __shared__ float uc[Lseq];   // u[:, h]   (gathered, row stride H)
    __shared__ float kc[Lseq];   // kern[h, :] (contiguous)
    const int h   = blockIdx.x;
    const int tid = threadIdx.x;

    if (tid < 32) {
        // kern row: 1 x 2048 contiguous tile
        tdm_load_f32_tile((uint32_t)(uintptr_t)kc, kernG + (size_t)h * Lseq,
                          /*tdim0=*/Lseq, /*tdim1=*/1,
                          /*tile0=*/Lseq, /*tile1=*/1, /*stride0=*/Lseq);
        // u column: 2048 rows x 1 element, row stride H -> compacted into LDS
        tdm_load_f32_tile((uint32_t)(uintptr_t)uc, u + h,
                          /*tdim0=*/1, /*tdim1=*/Lseq,
                          /*tile0=*/1, /*tile1=*/Lseq, /*stride0=*/Hch);
        __builtin_amdgcn_s_wait_tensorcnt(0);
    }
    __syncthreads();

    const int I     = tid >> 5;                 // tile index = wave id, 0..7
    const int lane  = tid & 31;
    const int halfl = lane >> 4;                // hi/lo half of the wave
    const int lm    = lane & 15;                // A row m / B col n

    v8f acc = (v8f){0.f, 0.f, 0.f, 0.f, 0.f, 0.f, 0.f, 0.f};

    // ---- d = 0 edge: kern index m-r can underflow; u index always valid ----
    {
        const int ub = 256 * I + 16 * lm;
        #pragma unroll
        for (int cch = 0; cch < 4; ++cch) {
            const int k0  = 4 * cch + 2 * halfl;
            const int ia0 = lm - k0;
            v2f a, b;
            a.x = (ia0 >= 0) ? kc[ia0] : 0.0f;
            a.y = (ia0 >= 1) ? kc[ia0 - 1] : 0.0f;
            b.x = uc[ub + k0];
            b.y = uc[ub + k0 + 1];
            acc = __builtin_amdgcn_wmma_f32_16x16x4_f32(
                false, a, false, b, (short)0, acc, false, false);
        }
    }
    // ---- interior: 1 <= d <= 16I, fully unmasked hot loop ----
    const int Dint = 16 * I;
    for (int d = 1; d <= Dint; ++d) {
        const int ka = 16 * d + lm;
        const int ub = 256 * I + 16 * lm - 16 * d;
        #pragma unroll
        for (int cch = 0; cch < 4; ++cch) {
            const int k0 = 4 * cch + 2 * halfl;
            v2f a, b;
            a.x = kc[ka - k0];
            a.y = kc[ka - k0 - 1];
            b.x = uc[ub + k0];
            b.y = uc[ub + k0 + 1];
            acc = __builtin_amdgcn_wmma_f32_16x16x4_f32(
                false, a, false, b, (short)0, acc, false, false);
        }
    }
    // ---- tail: 16I < d < 16I+16: u block can underflow (per-column mask) ----
    for (int d = Dint + 1; d < Dint + 16; ++d) {
        const int ka = 16 * d + lm;
        const int ub = 256 * I + 16 * lm - 16 * d;
        const bool on = (ub >= 0);              // column n=lm valid for this d
        #pragma unroll
        for (int cch = 0; cch < 4; ++cch) {
            const int k0 = 4 * cch + 2 * halfl;
            v2f a, b;
            a.x = kc[ka - k0];
            a.y = kc[ka - k0 - 1];
            b.x = on ? uc[ub + k0] : 0.0f;
            b.y = on ? uc[ub + k0 + 1] : 0.0f;
            acc = __builtin_amdgcn_wmma_f32_16x16x4_f32(
                false, a, false, b, (short)0, acc, false, false);
        }
    }

    // Epilogue: C/D layout — lane n = lm, row m = v + 8*halfl.
    const float dh = dvec[h];
    #pragma unroll
    for (int v = 0; v < 8; ++v) {
        const int m = v + 8 * halfl;
        const int t = 256 * I + 16 * lm + m;
        out[(size_t)t * Hch + h] = acc[v] + dh * uc[t];
    }
}

// ---------------------------------------------------------------------------
extern "C" void kernel_launch(void* const* d_in, const int* in_sizes, int n_in,
                              void* d_out, int out_size, void* d_ws, size_t ws_size,
                              hipStream_t stream)
{
    const float* u     = (const float*)d_in[0];
    const float* lamR  = (const float*)d_in[1];
    const float* lamI  = (const float*)d_in[2];
    const float* pR    = (const float*)d_in[3];
    const float* pI    = (const float*)d_in[4];
    const float* bR    = (const float*)d_in[5];
    const float* bI    = (const float*)d_in[6];
    const float* c     = (const float*)d_in[7];
    const float* dvec  = (const float*)d_in[8];
    const float* lstep = (const float*)d_in[9];

    float* kernG = (float*)d_ws;      // H * L floats = 4 MB scratch
    float* outp  = (float*)d_out;

    s4_kern_ifft<<<Hch, 256, 0, stream>>>(lamR, lamI, pR, pI, bR, bI, c,
                                          lstep, kernG);
    s4_conv_wmma<<<Hch, 256, 0, stream>>>(u, kernG, dvec, outp);
}